// MixHop_25555055411700
// MI455X (gfx1250) — compile-verified
//
#include <hip/hip_runtime.h>

// MixHop GCN for MI455X (gfx1250, wave32).
// Dense GEMMs: V_WMMA_F32_16X16X4_F32, 32x64 strip per wave (8 WMMA / k-step,
// B fragments reused across two M-tiles). Sparse propagation: float4-chunked
// device-scope relaxed f32 atomics (-> global_atomic_add_f32), L2-resident.

typedef float v2f __attribute__((ext_vector_type(2)));
typedef float v8f __attribute__((ext_vector_type(8)));

#define N_NODES 50000
#define N_EDGES 800000

static __device__ __forceinline__ void atomic_fadd(float* p, float v) {
    __hip_atomic_fetch_add(p, v, __ATOMIC_RELAXED, __HIP_MEMORY_SCOPE_AGENT);
}

// ---------------------------------------------------------------- norm stage

__global__ void k_deg_init(float* __restrict__ deg, int n) {
    int i = blockIdx.x * blockDim.x + threadIdx.x;
    if (i < n) deg[i] = 1.0f;                       // self-loop weight
}

__global__ void k_deg_accum(const int* __restrict__ col, const float* __restrict__ w,
                            float* __restrict__ deg, int e) {
    int i = blockIdx.x * blockDim.x + threadIdx.x;
    if (i < e) atomic_fadd(&deg[col[i]], w[i]);
}

__global__ void k_dinv(float* __restrict__ deg, int n) {
    int i = blockIdx.x * blockDim.x + threadIdx.x;
    if (i < n) { float d = deg[i]; deg[i] = d > 0.0f ? rsqrtf(d) : 0.0f; }
}

__global__ void k_norm(const int* __restrict__ row, const int* __restrict__ col,
                       const float* __restrict__ w, const float* __restrict__ dinv,
                       float* __restrict__ nrm, int e) {
    int i = blockIdx.x * blockDim.x + threadIdx.x;
    if (i < e) nrm[i] = dinv[row[i]] * w[i] * dinv[col[i]];
}

// ------------------------------------------------------------- propagation
// out = D^-1/2 (A + I) D^-1/2 h : init with self-loop term dinv^2 * h, then
// scatter-add norm[e]*h[row[e]] into out[col[e]], 4 channels per thread.

__global__ void k_prop_init(const float* __restrict__ h, const float* __restrict__ dinv,
                            float* __restrict__ out, int n, int C) {
    int i = blockIdx.x * blockDim.x + threadIdx.x;
    if (i < n * C) {
        int node = i / C;
        float d = dinv[node];
        out[i] = d * d * h[i];
    }
}

__global__ void k_prop_scatter(const int* __restrict__ row, const int* __restrict__ col,
                               const float* __restrict__ nrm, const float* __restrict__ h,
                               float* __restrict__ out, int e, int C) {
    int chunks = C >> 2;                            // float4 chunks per row
    int t = blockIdx.x * blockDim.x + threadIdx.x;
    if (t >= e * chunks) return;
    int ei = t / chunks;
    int c4 = (t - ei * chunks) << 2;
    float nm = nrm[ei];
    int r = row[ei];
    int c = col[ei];
    float4 hv = *(const float4*)(h + (size_t)r * C + c4);
    float* o = out + (size_t)c * C + c4;
    atomic_fadd(o + 0, nm * hv.x);
    atomic_fadd(o + 1, nm * hv.y);
    atomic_fadd(o + 2, nm * hv.z);
    atomic_fadd(o + 3, nm * hv.w);
}

// ------------------------------------------------------------------- GEMM
// out[n x 64] (+col offset, row stride `ostride`) = H[n x K] @ W[K x 64] + bias
// One wave32 computes a 32x64 strip: 2 M-tiles x 4 N-tiles, 8 accumulators.
// B fragments are loaded once per k-step and feed both M-tiles.
// A frag (16x4 f32): lane m=lane&15, kb=(lane>>4)*2 -> {H[m][k+kb], H[m][k+kb+1]}
// B frag (4x16 f32): v0 = W[k+kb][nn], v1 = W[k+kb+1][nn], nn = lane&15
// C/D: VGPR r -> rows r (lanes 0-15) / r+8 (lanes 16-31), col = lane&15

static __device__ __forceinline__ v8f wmma_f32x4(v2f a, v2f b, v8f c) {
    return __builtin_amdgcn_wmma_f32_16x16x4_f32(
        /*neg_a=*/false, a, /*neg_b=*/false, b,
        /*c_mod=*/(short)0, c, /*reuse_a=*/false, /*reuse_b=*/false);
}

template <int RELU>
__global__ void k_gemm_wmma(const float* __restrict__ H, const float* __restrict__ W,
                            const float* __restrict__ bias, float* __restrict__ out,
                            int n, int K, int ostride, int ocol) {
    int gtid = blockIdx.x * blockDim.x + threadIdx.x;
    int wave = gtid >> 5;                           // global wave index
    int lane = threadIdx.x & 31;
    int row0 = wave << 5;                           // 32 rows per wave
    if (row0 >= n) return;                          // wave-uniform
    bool two = (row0 + 16) < n;                     // wave-uniform tail handling

    int m  = lane & 15;
    int kb = (lane >> 4) << 1;

    v8f acc0 = {}, acc1 = {}, acc2 = {}, acc3 = {};
    v8f acc4 = {}, acc5 = {}, acc6 = {}, acc7 = {};

    const float* hp0 = H + (size_t)(row0 + m) * K + kb;
    // Tail wave: clamp second-tile stream to a valid address; results discarded.
    const float* hp1 = two ? (H + (size_t)(row0 + 16 + m) * K + kb) : hp0;
    const float* wp  = W + (size_t)kb * 64 + m;     // 64 = output width

    for (int k = 0; k < K; k += 4) {
        v2f a0 = { hp0[0],  hp0[1]       };
        v2f a1 = { hp1[0],  hp1[1]       };
        v2f b0 = { wp[0],   wp[64]       };
        v2f b1 = { wp[16],  wp[64 + 16]  };
        v2f b2 = { wp[32],  wp[64 + 32]  };
        v2f b3 = { wp[48],  wp[64 + 48]  };
        acc0 = wmma_f32x4(a0, b0, acc0);
        acc4 = wmma_f32x4(a1, b0, acc4);
        acc1 = wmma_f32x4(a0, b1, acc1);
        acc5 = wmma_f32x4(a1, b1, acc5);
        acc2 = wmma_f32x4(a0, b2, acc2);
        acc6 = wmma_f32x4(a1, b2, acc6);
        acc3 = wmma_f32x4(a0, b3, acc3);
        acc7 = wmma_f32x4(a1, b3, acc7);
        hp0 += 4;
        hp1 += 4;
        wp  += 4 * 64;
    }

    float bb0 = bias[m], bb1 = bias[16 + m], bb2 = bias[32 + m], bb3 = bias[48 + m];
    int rhalf = (lane >> 4) << 3;
#pragma unroll
    for (int r = 0; r < 8; ++r) {
        int rr = row0 + rhalf + r;
        float* op = out + (size_t)rr * ostride + ocol;
        float v0 = acc0[r] + bb0;
        float v1 = acc1[r] + bb1;
        float v2 = acc2[r] + bb2;
        float v3 = acc3[r] + bb3;
        if (RELU) {
            v0 = fmaxf(v0, 0.0f); v1 = fmaxf(v1, 0.0f);
            v2 = fmaxf(v2, 0.0f); v3 = fmaxf(v3, 0.0f);
        }
        op[m]      = v0;
        op[16 + m] = v1;
        op[32 + m] = v2;
        op[48 + m] = v3;
    }
    if (two) {
#pragma unroll
        for (int r = 0; r < 8; ++r) {
            int rr = row0 + 16 + rhalf + r;
            float* op = out + (size_t)rr * ostride + ocol;
            float v0 = acc4[r] + bb0;
            float v1 = acc5[r] + bb1;
            float v2 = acc6[r] + bb2;
            float v3 = acc7[r] + bb3;
            if (RELU) {
                v0 = fmaxf(v0, 0.0f); v1 = fmaxf(v1, 0.0f);
                v2 = fmaxf(v2, 0.0f); v3 = fmaxf(v3, 0.0f);
            }
            op[m]      = v0;
            op[16 + m] = v1;
            op[32 + m] = v2;
            op[48 + m] = v3;
        }
    }
}

// ---------------------------------------------------------------- launcher

extern "C" void kernel_launch(void* const* d_in, const int* in_sizes, int n_in,
                              void* d_out, int out_size, void* d_ws, size_t ws_size,
                              hipStream_t stream) {
    const float* x  = (const float*)d_in[0];   // [N,96]
    const int*   ei = (const int*)d_in[1];     // [2,E]
    const float* ew = (const float*)d_in[2];   // [E]
    const float* W1 = (const float*)d_in[3];   // [3,96,64]
    const float* b1 = (const float*)d_in[4];   // [3,64]
    const float* W2 = (const float*)d_in[5];   // [3,192,64]
    const float* b2 = (const float*)d_in[6];   // [3,64]
    const float* Wl = (const float*)d_in[7];   // [192,64]
    const float* bl = (const float*)d_in[8];   // [64]
    float* out = (float*)d_out;                // [N,64]

    const int N = N_NODES, E = N_EDGES;
    const int* row = ei;       // source
    const int* col = ei + E;   // target

    // Workspace layout (floats): dinv[N] | norm[E] | bufA[N*192] | bufB[N*192] | bufC[N*192]
    float* ws   = (float*)d_ws;
    float* dinv = ws;
    float* nrm  = dinv + N;
    float* bufA = nrm + E;
    float* bufB = bufA + (size_t)N * 192;
    float* bufC = bufB + (size_t)N * 192;

    float* p1 = bufA;                       // A x        [N,96]
    float* p2 = bufA + (size_t)N * 96;      // A^2 x      [N,96]
    float* h1 = bufB;                       // conv1 out  [N,192]
    float* q1 = bufA;                       // A h1       [N,192]  (p1/p2 dead)
    float* q2 = bufB;                       // A^2 h1     [N,192]  (h1 dead)
    float* h2 = bufC;                       // conv2 out  [N,192]

    dim3 B(256);
    auto g1 = [](int t) { return dim3((unsigned)((t + 255) / 256)); };

    // --- GCN symmetric normalization
    k_deg_init <<<g1(N), B, 0, stream>>>(dinv, N);
    k_deg_accum<<<g1(E), B, 0, stream>>>(col, ew, dinv, E);
    k_dinv     <<<g1(N), B, 0, stream>>>(dinv, N);
    k_norm     <<<g1(E), B, 0, stream>>>(row, col, ew, dinv, nrm, E);

    // --- conv1 propagations: p1 = A x ; p2 = A p1
    k_prop_init   <<<g1(N * 96), B, 0, stream>>>(x, dinv, p1, N, 96);
    k_prop_scatter<<<g1(E * 24), B, 0, stream>>>(row, col, nrm, x, p1, E, 96);
    k_prop_init   <<<g1(N * 96), B, 0, stream>>>(p1, dinv, p2, N, 96);
    k_prop_scatter<<<g1(E * 24), B, 0, stream>>>(row, col, nrm, p1, p2, E, 96);

    // --- conv1 GEMMs (concat into h1, fused bias+ReLU)
    int waves32 = (N + 31) / 32;             // 1563 strips (last is 16 rows)
    dim3 gg((unsigned)((waves32 + 3) / 4));  // 4 waves / 128-thread block
    dim3 gb(128);
    k_gemm_wmma<1><<<gg, gb, 0, stream>>>(x,  W1 + 0 * 96 * 64, b1 + 0,   h1, N, 96, 192, 0);
    k_gemm_wmma<1><<<gg, gb, 0, stream>>>(p1, W1 + 1 * 96 * 64, b1 + 64,  h1, N, 96, 192, 64);
    k_gemm_wmma<1><<<gg, gb, 0, stream>>>(p2, W1 + 2 * 96 * 64, b1 + 128, h1, N, 96, 192, 128);

    // --- conv2 propagations + GEMMs (careful buffer reuse ordering)
    // q1 = A h1  (into bufA; p1/p2 consumed above)
    k_prop_init   <<<g1(N * 192), B, 0, stream>>>(h1, dinv, q1, N, 192);
    k_prop_scatter<<<g1(E * 48),  B, 0, stream>>>(row, col, nrm, h1, q1, E, 192);
    // h2[:,0:64] = h1 @ W2[0]  (consume h1 before overwriting it with q2)
    k_gemm_wmma<0><<<gg, gb, 0, stream>>>(h1, W2 + 0 * 192 * 64, b2 + 0, h2, N, 192, 192, 0);
    // q2 = A q1  (into bufB; h1 dead)
    k_prop_init   <<<g1(N * 192), B, 0, stream>>>(q1, dinv, q2, N, 192);
    k_prop_scatter<<<g1(E * 48),  B, 0, stream>>>(row, col, nrm, q1, q2, E, 192);
    k_gemm_wmma<0><<<gg, gb, 0, stream>>>(q1, W2 + 1 * 192 * 64, b2 + 64,  h2, N, 192, 192, 64);
    k_gemm_wmma<0><<<gg, gb, 0, stream>>>(q2, W2 + 2 * 192 * 64, b2 + 128, h2, N, 192, 192, 128);

    // --- final linear: out = h2 @ Wl + bl
    k_gemm_wmma<0><<<gg, gb, 0, stream>>>(h2, Wl, bl, out, N, 192, 64, 0);
}